// ChebyshevBlock_28123445854756
// MI455X (gfx1250) — compile-verified
//
#include <hip/hip_runtime.h>
#include <hip/hip_bf16.h>
#include <math.h>
#include <stdint.h>

// Chebyshev positional embedding: out[pos][d] = T_d(x), x = 2*id/max_seq_len - 1.
//
// One wave32 per (b,s) position. Lane l owns the comb d = l + 32k via the
// shift identity T_{d+32} = 2*T_32*T_d - T_{d-32} (wave-uniform coefficient).
// Each 32-wide block is staged into a per-wave LDS tile (ds_store_b32), then
// the 4KB tile is drained with the CDNA5 async DMA path:
//   s_wait_dscnt 0  ->  8x global_store_async_from_lds_b128  ->  s_wait_asynccnt 0
// Each async instruction moves 32 lanes x 16B = 512B contiguous to global.

#define D_MODEL 1024
#define WAVE    32
#define DBLOCKS (D_MODEL / WAVE)   // 32 stride-32 blocks per position
#define WPB     8                  // waves per 256-thread block

__global__ __launch_bounds__(256)
void cheb_emb_kernel(const float* __restrict__ ids,
                     const int*   __restrict__ max_seq_len,
                     float*       __restrict__ out,
                     int n_pos) {
    __shared__ float tile[WPB * D_MODEL];              // 32 KB: 4KB per wave

    const int lane = threadIdx.x & (WAVE - 1);
    const int wv   = threadIdx.x >> 5;                 // wave id in block (wave32)
    const int pos  = blockIdx.x * (blockDim.x >> 5) + wv;
    if (pos >= n_pos) return;

    // Wave-uniform scalars. max_seq_len is a power of two -> v_rcp_f32 is exact.
    const float scale = 2.0f * __builtin_amdgcn_rcpf((float)(*max_seq_len));
    float x = fmaf(ids[pos], scale, -1.0f);
    x = fminf(fmaxf(x, -1.0f), 1.0f);                  // acos domain safety
    const float theta = acosf(x);

    // Closed-form seeds: T_l, T_{l+32}, and shift coefficient 2*T_32.
    float t_prev = __cosf((float)lane * theta);          // T_l        (d-block 0)
    float t_cur  = __cosf((float)(lane + WAVE) * theta); // T_{l+32}   (d-block 1)
    const float c2 = 2.0f * __cosf((float)WAVE * theta); // 2*T_32

    // Pin exact leading polynomials like the reference (T_0=1, T_1=x).
    if (lane == 0) t_prev = 1.0f;
    if (lane == 1) t_prev = x;

    // ---- Stage the 1024-float row into the per-wave LDS tile (ds_store_b32).
    float* __restrict__ row = &tile[wv * D_MODEL];
    row[0 * WAVE + lane] = t_prev;
    row[1 * WAVE + lane] = t_cur;
    #pragma unroll
    for (int k = 2; k < DBLOCKS; ++k) {
        const float t_next = fmaf(c2, t_cur, -t_prev);   // T_{d+32}=2*T_32*T_d-T_{d-32}
        row[k * WAVE + lane] = t_next;
        t_prev = t_cur;
        t_cur  = t_next;
    }

    // Make this wave's LDS writes visible before the async DMA reads them.
    asm volatile("s_wait_dscnt 0x0" ::: "memory");

    // ---- Async drain: LDS -> global, 16B per lane per op, 512B per wave-op.
    // Low 32 bits of a generic LDS pointer are the raw LDS byte address.
    uint32_t lds_addr = (uint32_t)(uintptr_t)&row[lane * 4];
    uint64_t gaddr    = (uint64_t)(uintptr_t)(out + (size_t)pos * D_MODEL)
                      + (uint64_t)(lane * 16);
    #pragma unroll
    for (int j = 0; j < D_MODEL * 4 / 512; ++j) {        // 8 chunks of 512B
        asm volatile("global_store_async_from_lds_b128 %0, %1, off"
                     :: "v"(gaddr), "v"(lds_addr)
                     : "memory");
        gaddr    += 512;
        lds_addr += 512;
    }

    // Explicit completion (s_endpgm would also imply wait-idle).
    asm volatile("s_wait_asynccnt 0x0" ::: "memory");
}

extern "C" void kernel_launch(void* const* d_in, const int* in_sizes, int n_in,
                              void* d_out, int out_size, void* d_ws, size_t ws_size,
                              hipStream_t stream) {
    const float* ids         = (const float*)d_in[0];   // [B*S] f32
    const int*   max_seq_len = (const int*)d_in[1];     // scalar on device
    // d_in[2] = d_model (scalar); fixed at 1024 per reference for full unroll.
    float* out = (float*)d_out;                          // [B*S, 1024] f32

    const int n_pos = in_sizes[0];                       // 32768
    const int grid  = (n_pos + WPB - 1) / WPB;

    cheb_emb_kernel<<<grid, 256, 0, stream>>>(ids, max_seq_len, out, n_pos);
}